// EdgeCreator_62904091018193
// MI455X (gfx1250) — compile-verified
//
#include <hip/hip_runtime.h>

// EdgeCreator for MI455X (gfx1250): gather + broadcast-subtract.
//   out[v, j, :] = feat[v, :] - (nidx[v, j+1] >= 0 ? feat[nidx[v, j+1], :] : 0)
// Roofline: 793.6 MB NT stores -> ~34 us HBM floor @ 23.3 TB/s; feat (25.6 MB)
// is L2-resident (192 MB L2) so the random gathers are L2-served; compute is
// ~198 MFLOP (irrelevant). No contraction dim => WMMA is not applicable; the
// CDNA5 paths that matter are async global->LDS DMA (ASYNCcnt), b128 vector
// memory, NT store hints, and global_prefetch.

typedef __attribute__((ext_vector_type(4))) float f32x4;

namespace {
constexpr int KNB  = 32;        // neighbors per vertex (col 0 = self)
constexpr int NEDG = KNB - 1;   // 31 output neighbors
constexpr int FDIM = 64;        // features
constexpr int VPB  = 4;         // vertices per block
constexpr int BLK  = 256;       // 8 waves (wave32)
}

__global__ __launch_bounds__(BLK) void edge_creator_kernel(
    const int*   __restrict__ nidx,
    const float* __restrict__ feat,
    float*       __restrict__ out,
    int V)
{
    __shared__ __align__(16) float s_self[VPB][FDIM]; // 4 x 256 B = 1 KB
    __shared__ __align__(16) int   s_idx [VPB][KNB];  // 4 x 128 B = 512 B

    const int t  = threadIdx.x;
    const int v0 = blockIdx.x * VPB;

    // ---- async DMA #1: 4 index rows (512 B) via wave 0, 32 lanes x b128 ----
    // gfx1250 GLOBAL_LOAD_ASYNC_TO_LDS_B128, GVS addressing:
    //   global address = 64-bit SGPR base + 32-bit VGPR offset; LDS address comes from the VDST VGPR
    if (t < 32) {
        const int vv = t >> 3;              // vertex in block
        const int c  = t & 7;               // 16-byte chunk within 128 B row
        int v = v0 + vv; if (v >= V) v = V - 1;
        unsigned ldsa = (unsigned)(size_t)&s_idx[vv][c * 4];
        unsigned goff = ((unsigned)v * (unsigned)KNB + (unsigned)c * 4u) * 4u;
        asm volatile("global_load_async_to_lds_b128 %0, %1, %2"
                     :: "v"(ldsa), "v"(goff), "s"(nidx)
                     : "memory");
    }

    // ---- async DMA #2: 4 "self" feature rows (1 KB) via waves 0-1 ----
    if (t < VPB * (FDIM / 4)) {             // 64 threads
        const int vv = t >> 4;
        const int c  = t & 15;
        int v = v0 + vv; if (v >= V) v = V - 1;
        unsigned ldsa = (unsigned)(size_t)&s_self[vv][c * 4];
        unsigned goff = ((unsigned)v * (unsigned)FDIM + (unsigned)c * 4u) * 4u;
        asm volatile("global_load_async_to_lds_b128 %0, %1, %2"
                     :: "v"(ldsa), "v"(goff), "s"(feat)
                     : "memory");
    }

    // warm L2 with the next block's index rows (one global_prefetch)
    if (t == 0) {
        int vn = v0 + VPB; if (vn >= V) vn = V - 1;
        __builtin_prefetch(&nidx[(size_t)vn * KNB], 0, 3);
    }

    asm volatile("s_wait_asynccnt 0" ::: "memory");
    __syncthreads();

    // ---- gather (L2-resident feat) + subtract + non-temporal b128 store ----
    const int c4 = (t & 15) * 4;   // float4 feature chunk
    const int g  = t >> 4;         // neighbor group 0..15
    const float* fchunk = feat + c4;

    for (int vv = 0; vv < VPB; ++vv) {
        const int v = v0 + vv;
        if (v >= V) break;

        const f32x4 self = *(const f32x4*)&s_self[vv][c4];
        float* orow = out + (size_t)v * NEDG * FDIM + c4;

        const int j0 = g;
        const int j1 = g + 16;
        const int i0 = s_idx[vv][1 + j0];
        const int i1 = (j1 < NEDG) ? s_idx[vv][1 + j1] : -1;

        f32x4 r0 = self;
        if (i0 >= 0) {
            const f32x4 nb = *(const f32x4*)(fchunk + (size_t)i0 * FDIM);
            r0 = self - nb;
        }
        f32x4 r1 = self;
        if (i1 >= 0) {
            const f32x4 nb = *(const f32x4*)(fchunk + (size_t)i1 * FDIM);
            r1 = self - nb;
        }

        __builtin_nontemporal_store(r0, (f32x4*)(orow + (size_t)j0 * FDIM));
        if (j1 < NEDG)
            __builtin_nontemporal_store(r1, (f32x4*)(orow + (size_t)j1 * FDIM));
    }
}

extern "C" void kernel_launch(void* const* d_in, const int* in_sizes, int n_in,
                              void* d_out, int out_size, void* d_ws, size_t ws_size,
                              hipStream_t stream) {
    const int*   nidx = (const int*)d_in[0];
    const float* feat = (const float*)d_in[1];
    float*       out  = (float*)d_out;
    (void)n_in; (void)out_size; (void)d_ws; (void)ws_size;

    const int V    = in_sizes[0] / KNB;        // 100000
    const int grid = (V + VPB - 1) / VPB;      // 25000 blocks x 256 threads
    edge_creator_kernel<<<grid, BLK, 0, stream>>>(nidx, feat, out, V);
}